// OneBlockHashModel_66623532695729
// MI455X (gfx1250) — compile-verified
//
#include <hip/hip_runtime.h>

// OneBlockHashModel on gfx1250 (MI455X), wave32.
// One wave processes a tile of 16 blocks. Dense layers computed as
// D^T = W x X^T with V_WMMA_F32_16X16X4_F32 (4 chained K-chunks, bias in C).
// The WMMA C/D layout equals the next layer's B layout up to a lane^16 swap
// (ds_swizzle SWAPX16 + cndmask). Tent map: 50 iters, register-resident,
// folded via f(x) = tent_left(min(x, 1-x)) -> 6 VALU ops/elem/iter.

typedef float v2f __attribute__((ext_vector_type(2)));
typedef float v4f __attribute__((ext_vector_type(4)));
typedef float v8f __attribute__((ext_vector_type(8)));

#define T_ITERS 50

__device__ __forceinline__ float swap16(float v) {
    // group-of-32 ds_swizzle, xor_mask=0x10, and_mask=0x1f  -> SWAPX16
    return __int_as_float(__builtin_amdgcn_ds_swizzle(__float_as_int(v), 0x401f));
}

// 50-step skew tent map on 8 register-resident values.
// Uses the reflection identity: f(x) = tent_left(min(x, 1-x)) where
// tent_left(z) = z<q ? z*invq : fma(z, invh, -q*invh).
__device__ __forceinline__ void tent50(float x[8], float q) {
    const float invq   = 1.0f / q;
    const float invh   = 1.0f / (0.5f - q);
    const float nqinvh = -q * invh;
    #pragma unroll 2
    for (int t = 0; t < T_ITERS; ++t) {
        #pragma unroll
        for (int r = 0; r < 8; ++r) {
            float v = x[r];
            float z = fminf(v, 1.0f - v);
            x[r] = (z < q) ? (z * invq) : __builtin_fmaf(z, invh, nqinvh);
        }
    }
}

// One dense layer: acc = bias; acc += A(=W chunks) x B(=X^T built from x/sx).
// x[] holds, per lane l: block m = l%16, feature n = r + 8*(l>=16).
__device__ __forceinline__ v8f dense16_wmma(const float x[8], const float a[4][2],
                                            const float bias[8], bool lo) {
    float sx[8];
    #pragma unroll
    for (int r = 0; r < 8; ++r) sx[r] = swap16(x[r]);

    v8f acc;
    #pragma unroll
    for (int r = 0; r < 8; ++r) acc[r] = bias[r];

    v2f A, Bv;
    // K-chunk 0: features 0..3
    A[0] = a[0][0]; A[1] = a[0][1];
    Bv[0] = lo ? x[0] : sx[2];  Bv[1] = lo ? x[1] : sx[3];
    acc = __builtin_amdgcn_wmma_f32_16x16x4_f32(false, A, false, Bv, (short)0, acc, false, false);
    // K-chunk 1: features 4..7
    A[0] = a[1][0]; A[1] = a[1][1];
    Bv[0] = lo ? x[4] : sx[6];  Bv[1] = lo ? x[5] : sx[7];
    acc = __builtin_amdgcn_wmma_f32_16x16x4_f32(false, A, false, Bv, (short)0, acc, false, false);
    // K-chunk 2: features 8..11
    A[0] = a[2][0]; A[1] = a[2][1];
    Bv[0] = lo ? sx[0] : x[2];  Bv[1] = lo ? sx[1] : x[3];
    acc = __builtin_amdgcn_wmma_f32_16x16x4_f32(false, A, false, Bv, (short)0, acc, false, false);
    // K-chunk 3: features 12..15
    A[0] = a[3][0]; A[1] = a[3][1];
    Bv[0] = lo ? sx[4] : x[6];  Bv[1] = lo ? sx[5] : x[7];
    acc = __builtin_amdgcn_wmma_f32_16x16x4_f32(false, A, false, Bv, (short)0, acc, false, false);
    return acc;
}

__global__ __launch_bounds__(256)
void hash_tiles_kernel(const float* __restrict__ in,
                       const float* __restrict__ W0, const float* __restrict__ B0,
                       const float* __restrict__ W1, const float* __restrict__ B1,
                       const float* __restrict__ W2, const float* __restrict__ B2,
                       const float* __restrict__ q0p, const float* __restrict__ q1p,
                       const float* __restrict__ q2p,
                       float* __restrict__ out, int ntiles) {
    const int lane = threadIdx.x & 31;
    const int wave = threadIdx.x >> 5;
    const int tile = blockIdx.x * (blockDim.x >> 5) + wave;   // wave-uniform
    if (tile >= ntiles) return;                               // whole wave exits -> EXEC stays full

    const int  m   = lane & 15;       // block index within tile (N of WMMA)
    const int  hi  = lane >> 4;       // lane-half selector
    const bool lo  = (hi == 0);
    const float q0 = q0p[0], q1 = q1p[0], q2 = q2p[0];

    // ---- wave-constant weights in A-matrix layout ----
    // A 16x4 f32 layout: lane m (0-15): VGPR0=K0, VGPR1=K1; lane m+16: K2, K3.
    float a1[4][2], a2[4][2];
    #pragma unroll
    for (int c = 0; c < 4; ++c) {
        #pragma unroll
        for (int j = 0; j < 2; ++j) {
            const int k = c * 4 + hi * 2 + j;
            a1[c][j] = W1[m * 16 + k];
            a2[c][j] = (m < 8) ? W2[m * 16 + k] : 0.0f;   // zero-pad rows 8..15
        }
    }
    float b1v[8], b2v[8];
    #pragma unroll
    for (int r = 0; r < 8; ++r) {
        const int n = r + 8 * hi;
        b1v[r] = B1[n];
        b2v[r] = (n < 8) ? B2[n] : 0.0f;
    }

    // ---- layerC: block-diagonal dot-4 + bias, directly in transposed layout ----
    // lane holds block m, features n = r + 8*hi  ->  reads input[m, 32*hi .. 32*hi+31]
    const float* src = in + (((size_t)tile * 16 + (size_t)m) * 64) + hi * 32;
    float x[8];
    #pragma unroll
    for (int r = 0; r < 8; ++r) {
        const int n = r + 8 * hi;
        v4f iv = *(const v4f*)(src + r * 4);
        v4f wv = *(const v4f*)(W0 + n * 4);
        float acc = B0[n] + iv.x * wv.x + iv.y * wv.y + iv.z * wv.z + iv.w * wv.w;
        x[r] = acc - floorf(acc);                       // mod 1
    }
    tent50(x, q0);

    // ---- layerD: 16x16 dense via WMMA, then mod + tent ----
    {
        v8f acc = dense16_wmma(x, a1, b1v, lo);
        #pragma unroll
        for (int r = 0; r < 8; ++r) { float v = acc[r]; x[r] = v - floorf(v); }
    }
    tent50(x, q1);

    // ---- layerH: 8x16 dense (zero-padded to 16) via WMMA, mod + tent ----
    {
        v8f acc = dense16_wmma(x, a2, b2v, lo);
        #pragma unroll
        for (int r = 0; r < 8; ++r) { float v = acc[r]; x[r] = v - floorf(v); }
    }
    tent50(x, q2);

    // ---- store: features 0..7 of block m live in lanes 0..15, VGPRs 0..7 ----
    if (lo) {
        float* dst = out + ((size_t)tile * 16 + (size_t)m) * 8;
        v4f o0 = { x[0], x[1], x[2], x[3] };
        v4f o1 = { x[4], x[5], x[6], x[7] };
        *(v4f*)(dst)     = o0;
        *(v4f*)(dst + 4) = o1;
    }
}

extern "C" void kernel_launch(void* const* d_in, const int* in_sizes, int n_in,
                              void* d_out, int out_size, void* d_ws, size_t ws_size,
                              hipStream_t stream) {
    const float* in = (const float*)d_in[0];
    const float* W0 = (const float*)d_in[1];
    const float* B0 = (const float*)d_in[2];
    const float* W1 = (const float*)d_in[3];
    const float* B1 = (const float*)d_in[4];
    const float* W2 = (const float*)d_in[5];
    const float* B2 = (const float*)d_in[6];
    const float* q0 = (const float*)d_in[7];
    const float* q1 = (const float*)d_in[8];
    const float* q2 = (const float*)d_in[9];
    float* out = (float*)d_out;

    const int nblocks = in_sizes[0] / 64;          // 2^21
    const int ntiles  = (nblocks + 15) / 16;       // 16 blocks per wave
    const int threads = 256;                       // 8 waves per workgroup
    const int waves_per_wg = threads / 32;
    const int grid = (ntiles + waves_per_wg - 1) / waves_per_wg;

    hash_tiles_kernel<<<grid, threads, 0, stream>>>(in, W0, B0, W1, B1, W2, B2,
                                                    q0, q1, q2, out, ntiles);
}